// ColorLoss_29609504539431
// MI455X (gfx1250) — compile-verified
//
#include <hip/hip_runtime.h>
#include <hip/hip_bf16.h>

// Problem constants (from reference): B=32, C=18, H=W=256, PATCH=15, PAD=7
#define B_      32
#define C_      18
#define W_      256
#define HW      65536          // 256*256
#define PATCH   15
#define PAD     7
#define NPAIR   (B_ * C_)      // 576
#define NELEM   (NPAIR * PATCH * PATCH * 3)   // 388800

typedef __attribute__((ext_vector_type(2))) float v2f;
typedef __attribute__((ext_vector_type(4))) float v4f;
typedef __attribute__((ext_vector_type(8))) float v8f;

// ---------------------------------------------------------------------------
// Kernel 1: per-(b,c) argmax over a 256x256 heatmap.
// Grid: 2*NPAIR blocks (first NPAIR -> bp_in, rest -> bp_out), 256 threads.
// Streaming scan with 128-bit non-temporal loads (302 MB total > 192 MB L2).
// Tie-break: lowest flat index wins (matches jnp.argmax first-occurrence).
// ---------------------------------------------------------------------------
__global__ __launch_bounds__(256) void argmax_kernel(const float* __restrict__ bp_in,
                                                     const float* __restrict__ bp_out,
                                                     float* __restrict__ best_val,
                                                     int* __restrict__ best_idx) {
    const int p = blockIdx.x;                       // [0, 2*NPAIR)
    const float* src = (p < NPAIR) ? (bp_in + (size_t)p * HW)
                                   : (bp_out + (size_t)(p - NPAIR) * HW);
    const int t = threadIdx.x;

    const v4f* src4 = (const v4f*)src;
    float bv = -1.0f;                               // heatmap values in [0,1)
    int   bi = 0;

    // 65536 floats = 16384 v4f = 64 per thread, coalesced stride-256
    #pragma unroll 4
    for (int i = 0; i < HW / 4 / 256; ++i) {
        const int e = i * 256 + t;                  // strictly increasing per thread
        v4f v = __builtin_nontemporal_load(&src4[e]);
        const int base = e * 4;
        if (v.x > bv) { bv = v.x; bi = base;     }
        if (v.y > bv) { bv = v.y; bi = base + 1; }
        if (v.z > bv) { bv = v.z; bi = base + 2; }
        if (v.w > bv) { bv = v.w; bi = base + 3; }
    }

    __shared__ float sval[256];
    __shared__ int   sidx[256];
    sval[t] = bv;
    sidx[t] = bi;
    __syncthreads();

    #pragma unroll
    for (int stride = 128; stride > 0; stride >>= 1) {
        if (t < stride) {
            const float ov = sval[t + stride];
            const int   oi = sidx[t + stride];
            if (ov > sval[t] || (ov == sval[t] && oi < sidx[t])) {
                sval[t] = ov;
                sidx[t] = oi;
            }
        }
        __syncthreads();
    }
    if (t == 0) {
        best_val[p] = sval[0];
        best_idx[p] = sidx[0];
    }
}

// ---------------------------------------------------------------------------
// Kernel 2: gather 15x15x3 patches around both keypoints, squared-diff sum.
// Grid: NPAIR blocks, 256 threads (225 active, one per patch position).
// Out-of-bounds pixels read as -1.0 (reference pads with constant -1).
// Invisible pairs (max <= 0.5 in either map) contribute exactly 0.
// ---------------------------------------------------------------------------
__global__ __launch_bounds__(256) void patch_mse_kernel(const float* __restrict__ img_in,
                                                        const float* __restrict__ img_out,
                                                        const float* __restrict__ best_val,
                                                        const int* __restrict__ best_idx,
                                                        float* __restrict__ partial) {
    const int p = blockIdx.x;                       // [0, NPAIR)
    const int t = threadIdx.x;

    const float m1 = best_val[p];
    const float m2 = best_val[NPAIR + p];
    if (!(m1 > 0.5f && m2 > 0.5f)) {                // block-uniform branch
        if (t == 0) partial[p] = 0.0f;
        return;
    }

    const int i1 = best_idx[p];
    const int i2 = best_idx[NPAIR + p];
    const int y1 = i1 >> 8, x1 = i1 & 255;
    const int y2 = i2 >> 8, x2 = i2 & 255;

    const int b = p / C_;
    const float* base1 = img_in  + (size_t)b * 3 * HW;
    const float* base2 = img_out + (size_t)b * 3 * HW;

    float acc = 0.0f;
    if (t < PATCH * PATCH) {
        const int dy = t / PATCH, dx = t % PATCH;
        const int yy1 = y1 + dy - PAD, xx1 = x1 + dx - PAD;
        const int yy2 = y2 + dy - PAD, xx2 = x2 + dx - PAD;
        const bool in1 = ((unsigned)yy1 < 256u) && ((unsigned)xx1 < 256u);
        const bool in2 = ((unsigned)yy2 < 256u) && ((unsigned)xx2 < 256u);
        const int o1 = yy1 * W_ + xx1;
        const int o2 = yy2 * W_ + xx2;
        #pragma unroll
        for (int ch = 0; ch < 3; ++ch) {
            const float g1 = in1 ? base1[ch * HW + o1] : -1.0f;
            const float g2 = in2 ? base2[ch * HW + o2] : -1.0f;
            const float d = g2 - g1;
            acc = fmaf(d, d, acc);
        }
    }

    __shared__ float s[256];
    s[t] = acc;
    __syncthreads();
    #pragma unroll
    for (int stride = 128; stride > 0; stride >>= 1) {
        if (t < stride) s[t] += s[t + stride];
        __syncthreads();
    }
    if (t == 0) partial[p] = s[0];
}

// ---------------------------------------------------------------------------
// Kernel 3: final reduction of 576 partials via V_WMMA_F32_16X16X4_F32.
// One wave32, EXEC all-1s. B = all-ones, so D[m,n] = sum_k A[m,k] + C[m,n],
// exact in f32. 9 accumulating WMMAs cover 9*64 = 576 values. Row sums land
// replicated across columns; lane0 VGPRs hold rows 0..7, lane16 rows 8..15.
// ---------------------------------------------------------------------------
__global__ __launch_bounds__(32) void final_reduce_wmma(const float* __restrict__ partial,
                                                        float* __restrict__ out) {
    const int lane = threadIdx.x;                   // 32 threads exactly

    v8f acc = {0.f, 0.f, 0.f, 0.f, 0.f, 0.f, 0.f, 0.f};
    v2f ones;
    ones.x = 1.0f;
    ones.y = 1.0f;

    #pragma unroll
    for (int base = 0; base < NPAIR; base += 64) {  // 576 / 64 = 9 iters, no tail
        v2f a;
        a.x = partial[base + lane * 2];
        a.y = partial[base + lane * 2 + 1];
        // 8 args: (neg_a, A, neg_b, B, c_mod, C, reuse_a, reuse_b)
        acc = __builtin_amdgcn_wmma_f32_16x16x4_f32(
            /*neg_a=*/false, a, /*neg_b=*/false, ones,
            /*c_mod=*/(short)0, acc, /*reuse_a=*/false, /*reuse_b=*/false);
    }

    // Sum the 8 row-sums held by this lane (column 0 view for lanes 0 and 16).
    float s = acc[0] + acc[1] + acc[2] + acc[3] + acc[4] + acc[5] + acc[6] + acc[7];

    __shared__ float red[32];
    red[lane] = s;
    __syncthreads();
    if (lane == 0) {
        const float scale = 1.0f / (float)NELEM;    // LAMBDA_PATCH = 1.0
        out[0] = (red[0] + red[16]) * scale;        // rows 0..7 + rows 8..15
    }
}

// ---------------------------------------------------------------------------
// Launch. Workspace layout: best_val[2*576] f32 | best_idx[2*576] i32 |
// partial[576] f32  (= 11.25 KB total).
// ---------------------------------------------------------------------------
extern "C" void kernel_launch(void* const* d_in, const int* in_sizes, int n_in,
                              void* d_out, int out_size, void* d_ws, size_t ws_size,
                              hipStream_t stream) {
    const float* img_in  = (const float*)d_in[0];
    const float* bp_in   = (const float*)d_in[1];
    const float* img_out = (const float*)d_in[2];
    const float* bp_out  = (const float*)d_in[3];
    float* out = (float*)d_out;

    float* best_val = (float*)d_ws;                  // 2*NPAIR floats
    int*   best_idx = (int*)(best_val + 2 * NPAIR);  // 2*NPAIR ints
    float* partial  = (float*)(best_idx + 2 * NPAIR); // NPAIR floats

    argmax_kernel<<<2 * NPAIR, 256, 0, stream>>>(bp_in, bp_out, best_val, best_idx);
    patch_mse_kernel<<<NPAIR, 256, 0, stream>>>(img_in, img_out, best_val, best_idx, partial);
    final_reduce_wmma<<<1, 32, 0, stream>>>(partial, out);
}